// hk3dlayer_srh_31301721653366
// MI455X (gfx1250) — compile-verified
//
#include <hip/hip_runtime.h>
#include <hip/hip_bf16.h>

typedef float v2f __attribute__((ext_vector_type(2)));
typedef float v8f __attribute__((ext_vector_type(8)));

#define HW 16384
#define IMG 128

__device__ __forceinline__ v8f wmma_f32_4(v2f a, v2f b, v8f c) {
    // D = A(16x4) * B(4x16) + C, fp32 WMMA (8-arg form: neg_a,A,neg_b,B,c_mod,C,reuse_a,reuse_b)
    return __builtin_amdgcn_wmma_f32_16x16x4_f32(false, a, false, b, (short)0, c, false, false);
}

// ---------------------------------------------------------------------------
// K1: conv1 1x1 GEMM  h1pre[b,o,p] = sum_c x[b,c,p]*w[o,c] + bias[o]
//     M=64(o) K=256(c) N=16384*16(p).  Block: 64o x 128p, wave: 16o x 64p.
// ---------------------------------------------------------------------------
__global__ __launch_bounds__(256) void gemm1_kernel(
    const float* __restrict__ x, const float* __restrict__ w,
    const float* __restrict__ bias, float* __restrict__ out) {
    const int wv = threadIdx.x >> 5;
    const int lane = threadIdx.x & 31;
    const int lh = lane >> 4;            // half-wave: selects K pair
    const int l = lane & 15;
    const int b = blockIdx.x >> 7;       // 128 p-tiles per batch image
    const int p0 = ((blockIdx.x & 127) << 7) + ((wv >> 2) << 6);
    const int otile = (wv & 3) << 4;

    v8f acc0 = {}, acc1 = {}, acc2 = {}, acc3 = {};
    const float* xb = x + (size_t)b * 256 * HW + p0 + l;
    const float* wrow = w + (otile + l) * 256;

    for (int k4 = 0; k4 < 256; k4 += 4) {
        const int kk = k4 + 2 * lh;
        v2f a = *(const v2f*)(wrow + kk);               // w[o, kk], w[o, kk+1]
        const float* xc0 = xb + (size_t)kk * HW;
        const float* xc1 = xc0 + HW;
        v2f b0, b1, b2, b3;
        b0.x = xc0[0];  b0.y = xc1[0];
        b1.x = xc0[16]; b1.y = xc1[16];
        b2.x = xc0[32]; b2.y = xc1[32];
        b3.x = xc0[48]; b3.y = xc1[48];
        acc0 = wmma_f32_4(a, b0, acc0);
        acc1 = wmma_f32_4(a, b1, acc1);
        acc2 = wmma_f32_4(a, b2, acc2);
        acc3 = wmma_f32_4(a, b3, acc3);
    }
    #pragma unroll
    for (int j = 0; j < 8; ++j) {
        const int o = otile + j + (lh << 3);
        const float bv = bias[o];
        float* op = out + ((size_t)b * 64 + o) * HW + p0 + l;
        op[0]  = acc0[j] + bv;
        op[16] = acc1[j] + bv;
        op[32] = acc2[j] + bv;
        op[48] = acc3[j] + bv;
    }
}

// ---------------------------------------------------------------------------
// GroupNorm stats -> per-(b,ch) scale/shift:  y = x*scale + shift
// one block per (b, group)
// ---------------------------------------------------------------------------
__global__ __launch_bounds__(256) void gn_stats_kernel(
    const float* __restrict__ src, const float* __restrict__ gamma,
    const float* __restrict__ beta, float* __restrict__ scale,
    float* __restrict__ shift, int C, int chPerGroup) {
    __shared__ float s1[256], s2[256];
    const int numGroups = C / chPerGroup;
    const int b = blockIdx.x / numGroups;
    const int g = blockIdx.x % numGroups;
    const size_t base = ((size_t)b * C + (size_t)g * chPerGroup) * HW;
    const int n = chPerGroup * HW;
    float a1 = 0.f, a2 = 0.f;
    for (int i = threadIdx.x; i < n; i += 256) {
        float v = src[base + i];
        a1 += v; a2 += v * v;
    }
    s1[threadIdx.x] = a1; s2[threadIdx.x] = a2;
    __syncthreads();
    for (int st = 128; st > 0; st >>= 1) {
        if ((int)threadIdx.x < st) {
            s1[threadIdx.x] += s1[threadIdx.x + st];
            s2[threadIdx.x] += s2[threadIdx.x + st];
        }
        __syncthreads();
    }
    if (threadIdx.x == 0) {
        const float inv_n = 1.f / (float)n;
        const float mean = s1[0] * inv_n;
        const float var = s2[0] * inv_n - mean * mean;
        const float rstd = rsqrtf(var + 1e-5f);
        for (int cc = 0; cc < chPerGroup; ++cc) {
            const int ch = g * chPerGroup + cc;
            const float sc = rstd * gamma[ch];
            scale[b * C + ch] = sc;
            shift[b * C + ch] = beta[ch] - mean * sc;
        }
    }
}

// ---------------------------------------------------------------------------
// K3: effective kernels.  band index bi = max(chebyshev_dist - 1, 0)
// factor(pos) = suffix_sum of softmax weights from band bi
// ---------------------------------------------------------------------------
__global__ __launch_bounds__(256) void eff_weights_kernel(
    const float* __restrict__ W1, const float* __restrict__ W2,
    float* __restrict__ W1e, float* __restrict__ W2e) {
    __shared__ float pnum[4][256];
    __shared__ float suf2[4];
    const int tid = threadIdx.x;
    float n0 = 0.f, n1 = 0.f, n2 = 0.f, n3 = 0.f;
    for (int e = tid; e < 5184; e += 256) {
        const int uv = e % 81;
        const int u = uv / 9, v = uv % 9;
        const int d = max(abs(u - 4), abs(v - 4));
        const int bi = max(d - 1, 0);
        const float val = W2[e];
        if (bi == 0) n0 += val; else if (bi == 1) n1 += val;
        else if (bi == 2) n2 += val; else n3 += val;
    }
    pnum[0][tid] = n0; pnum[1][tid] = n1; pnum[2][tid] = n2; pnum[3][tid] = n3;
    __syncthreads();
    for (int st = 128; st > 0; st >>= 1) {
        if (tid < st) {
            #pragma unroll
            for (int i = 0; i < 4; ++i) pnum[i][tid] += pnum[i][tid + st];
        }
        __syncthreads();
    }
    if (tid == 0) {
        // --- 2D softmax over bands ---
        const float cnt2[4] = {9.f, 16.f, 24.f, 32.f};
        float lg[4], mx = -1e30f;
        #pragma unroll
        for (int i = 0; i < 4; ++i) {
            lg[i] = pnum[i][0] / (64.f * cnt2[i] + 1e-10f);
            mx = fmaxf(mx, lg[i]);
        }
        float ex[4], se = 0.f;
        #pragma unroll
        for (int i = 0; i < 4; ++i) { ex[i] = expf(lg[i] - mx); se += ex[i]; }
        float a2[4];
        #pragma unroll
        for (int i = 0; i < 4; ++i) a2[i] = ex[i] / se;
        suf2[3] = a2[3];
        for (int i = 2; i >= 0; --i) suf2[i] = suf2[i + 1] + a2[i];
        // --- 1D kernel (9 taps, serial) ---
        const float cnt1[4] = {3.f, 2.f, 2.f, 2.f};
        float num1[4] = {0.f, 0.f, 0.f, 0.f};
        for (int j = 0; j < 9; ++j) num1[max(abs(j - 4) - 1, 0)] += W1[j];
        float lg1[4]; mx = -1e30f;
        #pragma unroll
        for (int i = 0; i < 4; ++i) {
            lg1[i] = num1[i] / (cnt1[i] + 1e-10f);
            mx = fmaxf(mx, lg1[i]);
        }
        se = 0.f;
        #pragma unroll
        for (int i = 0; i < 4; ++i) { ex[i] = expf(lg1[i] - mx); se += ex[i]; }
        float a1[4];
        #pragma unroll
        for (int i = 0; i < 4; ++i) a1[i] = ex[i] / se;
        float suf1[4];
        suf1[3] = a1[3];
        for (int i = 2; i >= 0; --i) suf1[i] = suf1[i + 1] + a1[i];
        for (int j = 0; j < 9; ++j)
            W1e[j] = W1[j] * suf1[max(abs(j - 4) - 1, 0)];
    }
    __syncthreads();
    for (int e = tid; e < 5184; e += 256) {
        const int uv = e % 81;
        const int bi = max(max(abs(uv / 9 - 4), abs(uv % 9 - 4)) - 1, 0);
        W2e[e] = W2[e] * suf2[bi];
    }
}

// ---------------------------------------------------------------------------
// K4: s = (9-tap channel conv) + (9x9 depthwise spatial conv) of normalized h1
//     16x16 output tile; 24x24 haloed LDS tile for the spatial part.
// ---------------------------------------------------------------------------
__global__ __launch_bounds__(256) void mixconv_kernel(
    const float* __restrict__ h1pre,
    const float* __restrict__ scaleA, const float* __restrict__ shiftA,
    const float* __restrict__ W1e, const float* __restrict__ W2e,
    float* __restrict__ s) {
    __shared__ float tile[24 * 24];
    __shared__ float w2s[81];
    __shared__ float w1s[9];
    const int blk = blockIdx.x;
    const int t = blk & 63;                 // 8x8 tiles of 16x16
    const int ch = (blk >> 6) & 63;
    const int b = blk >> 12;
    const int h0 = (t >> 3) << 4, w0 = (t & 7) << 4;
    const int tid = threadIdx.x;
    const float sc = scaleA[b * 64 + ch];
    const float sh = shiftA[b * 64 + ch];
    const float* plane = h1pre + ((size_t)b * 64 + ch) * HW;
    for (int i = tid; i < 576; i += 256) {
        const int r = i / 24, c = i % 24;
        const int gh = h0 + r - 4, gw = w0 + c - 4;
        float v = 0.f;
        if (gh >= 0 && gh < IMG && gw >= 0 && gw < IMG)
            v = plane[gh * IMG + gw] * sc + sh;
        tile[i] = v;
    }
    if (tid < 81) w2s[tid] = W2e[ch * 81 + tid];
    if (tid < 9)  w1s[tid] = W1e[tid];
    __syncthreads();
    const int ty = tid >> 4, tx = tid & 15;
    float acc = 0.f;
    #pragma unroll
    for (int u = 0; u < 9; ++u) {
        #pragma unroll
        for (int v = 0; v < 9; ++v)
            acc += w2s[u * 9 + v] * tile[(ty + u) * 24 + tx + v];
    }
    const int pix = (h0 + ty) * IMG + (w0 + tx);
    #pragma unroll
    for (int k = 0; k < 9; ++k) {
        const int cc = ch + k - 4;
        if (cc >= 0 && cc < 64) {
            const float v = h1pre[((size_t)b * 64 + cc) * HW + pix] *
                            scaleA[b * 64 + cc] + shiftA[b * 64 + cc];
            acc += w1s[k] * v;
        }
    }
    s[((size_t)b * 64 + ch) * HW + pix] = acc;
}

// ---------------------------------------------------------------------------
// K6: conv2 1x1 GEMM  out[b,o,p] = sum_ch relu(s*scale+shift)[b,ch,p]*w[o,ch]+bias
//     M=256(o) K=64(ch).  Block: 64o x 128p, wave: 16o x 64p.
// ---------------------------------------------------------------------------
__global__ __launch_bounds__(256) void gemm2_kernel(
    const float* __restrict__ s, const float* __restrict__ w,
    const float* __restrict__ bias, const float* __restrict__ scaleS,
    const float* __restrict__ shiftS, float* __restrict__ out) {
    const int blk = blockIdx.x;              // b(16) * og(4) * pt(128)
    const int b = blk >> 9;
    const int og = (blk >> 7) & 3;
    const int wv = threadIdx.x >> 5;
    const int lane = threadIdx.x & 31;
    const int lh = lane >> 4;
    const int l = lane & 15;
    const int otile = (og << 6) + ((wv & 3) << 4);
    const int p0 = ((blk & 127) << 7) + ((wv >> 2) << 6);

    v8f acc0 = {}, acc1 = {}, acc2 = {}, acc3 = {};
    const float* sb = s + (size_t)b * 64 * HW + p0 + l;
    const float* scb = scaleS + b * 64;
    const float* shb = shiftS + b * 64;
    const float* wrow = w + (otile + l) * 64;

    for (int k4 = 0; k4 < 64; k4 += 4) {
        const int kk = k4 + 2 * lh;
        v2f a = *(const v2f*)(wrow + kk);
        const float sc0 = scb[kk], sh0 = shb[kk];
        const float sc1 = scb[kk + 1], sh1 = shb[kk + 1];
        const float* sp0 = sb + (size_t)kk * HW;
        const float* sp1 = sp0 + HW;
        v2f b0, b1, b2, b3;
        b0.x = fmaxf(sp0[0]  * sc0 + sh0, 0.f); b0.y = fmaxf(sp1[0]  * sc1 + sh1, 0.f);
        b1.x = fmaxf(sp0[16] * sc0 + sh0, 0.f); b1.y = fmaxf(sp1[16] * sc1 + sh1, 0.f);
        b2.x = fmaxf(sp0[32] * sc0 + sh0, 0.f); b2.y = fmaxf(sp1[32] * sc1 + sh1, 0.f);
        b3.x = fmaxf(sp0[48] * sc0 + sh0, 0.f); b3.y = fmaxf(sp1[48] * sc1 + sh1, 0.f);
        acc0 = wmma_f32_4(a, b0, acc0);
        acc1 = wmma_f32_4(a, b1, acc1);
        acc2 = wmma_f32_4(a, b2, acc2);
        acc3 = wmma_f32_4(a, b3, acc3);
    }
    #pragma unroll
    for (int j = 0; j < 8; ++j) {
        const int o = otile + j + (lh << 3);
        const float bv = bias[o];
        float* op = out + ((size_t)b * 256 + o) * HW + p0 + l;
        op[0]  = acc0[j] + bv;
        op[16] = acc1[j] + bv;
        op[32] = acc2[j] + bv;
        op[48] = acc3[j] + bv;
    }
}

// ---------------------------------------------------------------------------
// K8: out = relu(out*scale + shift) + x   (in place, float4)
// ---------------------------------------------------------------------------
__global__ __launch_bounds__(256) void finalize_kernel(
    float* __restrict__ out, const float* __restrict__ x,
    const float* __restrict__ scale2, const float* __restrict__ shift2) {
    const int total4 = (16 * 256 * HW) / 4;
    for (int i = blockIdx.x * blockDim.x + threadIdx.x; i < total4;
         i += gridDim.x * blockDim.x) {
        const int row = (i << 2) / HW;           // b*256 + ch
        const float sc = scale2[row], sh = shift2[row];
        const float4 v = ((const float4*)out)[i];
        const float4 xv = ((const float4*)x)[i];
        float4 r;
        r.x = fmaxf(v.x * sc + sh, 0.f) + xv.x;
        r.y = fmaxf(v.y * sc + sh, 0.f) + xv.y;
        r.z = fmaxf(v.z * sc + sh, 0.f) + xv.z;
        r.w = fmaxf(v.w * sc + sh, 0.f) + xv.w;
        ((float4*)out)[i] = r;
    }
}

extern "C" void kernel_launch(void* const* d_in, const int* in_sizes, int n_in,
                              void* d_out, int out_size, void* d_ws, size_t ws_size,
                              hipStream_t stream) {
    const float* x    = (const float*)d_in[0];
    const float* W1   = (const float*)d_in[1];   // 9
    const float* W2   = (const float*)d_in[2];   // 64*81
    const float* c1w  = (const float*)d_in[3];   // 64x256
    const float* c1b  = (const float*)d_in[4];
    const float* c2w  = (const float*)d_in[5];   // 256x64
    const float* c2b  = (const float*)d_in[6];
    const float* gn1w = (const float*)d_in[7];
    const float* gn1b = (const float*)d_in[8];
    const float* gnsw = (const float*)d_in[9];
    const float* gnsb = (const float*)d_in[10];
    const float* gn2w = (const float*)d_in[11];
    const float* gn2b = (const float*)d_in[12];
    float* out = (float*)d_out;

    float* ws = (float*)d_ws;
    float* h1pre  = ws;                       // 16*64*16384
    float* s      = ws + 16777216;            // 16*64*16384
    float* scaleA = ws + 2 * 16777216;        // 16*64
    float* shiftA = scaleA + 1024;
    float* scaleS = shiftA + 1024;
    float* shiftS = scaleS + 1024;
    float* scale2 = shiftS + 1024;            // 16*256
    float* shift2 = scale2 + 4096;
    float* W1e    = shift2 + 4096;            // 16 (9 used)
    float* W2e    = W1e + 16;                 // 5184

    gemm1_kernel<<<2048, 256, 0, stream>>>(x, c1w, c1b, h1pre);
    gn_stats_kernel<<<256, 256, 0, stream>>>(h1pre, gn1w, gn1b, scaleA, shiftA, 64, 4);
    eff_weights_kernel<<<1, 256, 0, stream>>>(W1, W2, W1e, W2e);
    mixconv_kernel<<<65536, 256, 0, stream>>>(h1pre, scaleA, shiftA, W1e, W2e, s);
    gn_stats_kernel<<<256, 256, 0, stream>>>(s, gnsw, gnsb, scaleS, shiftS, 64, 4);
    gemm2_kernel<<<8192, 256, 0, stream>>>(s, c2w, c2b, scaleS, shiftS, out);
    gn_stats_kernel<<<256, 256, 0, stream>>>(out, gn2w, gn2b, scale2, shift2, 256, 16);
    finalize_kernel<<<4096, 256, 0, stream>>>(out, x, scale2, shift2);
}